// AdditiveAttention_71571335020797
// MI455X (gfx1250) — compile-verified
//
#include <hip/hip_runtime.h>
#include <hip/hip_bf16.h>

typedef __attribute__((ext_vector_type(8)))  float    v8f;
typedef __attribute__((ext_vector_type(2)))  float    v2f;

#define NSQ 512
#define BB 2
#define DMODEL 512
#define NHH 8
#define DHH 64
#define NSB (NSQ*BB)                    // 1024 rows of fused GEMM
#define NOUT (NHH*DHH)                  // 512 cols
#define ATTN_ELEMS ((size_t)BB*NHH*NSQ*NSQ)   // 4194304
#define KOUT_ELEMS ((size_t)BB*NHH*NSQ*DHH)   // 524288

// ---- CDNA5 async memory->LDS ops (ASYNCcnt-tracked, no VGPR round-trip) ----
__device__ __forceinline__ void async_ld_b128(unsigned lds_off, const float* g) {
  asm volatile("global_load_async_to_lds_b128 %0, %1, off"
               :: "v"(lds_off), "v"((unsigned long long)(uintptr_t)g) : "memory");
}
__device__ __forceinline__ void async_ld_b32(unsigned lds_off, const float* g) {
  asm volatile("global_load_async_to_lds_b32 %0, %1, off"
               :: "v"(lds_off), "v"((unsigned long long)(uintptr_t)g) : "memory");
}
__device__ __forceinline__ unsigned lds_addr(const void* p) {
  return (unsigned)(uintptr_t)p;   // flat addr[31:0] == wave-relative LDS addr
}

// ---------------------------------------------------------------------------
// Kernel A: fused projections q = query@Wq + bq, k = query@Wk + bk using the
// EXACT-f32 matrix core op V_WMMA_F32_16X16X4_F32 (bit-compatible with the
// f32 reference). 128 threads = 4 waves; tile 32(M) x 32(N); each wave owns
// one 16x16 tile for BOTH weight matrices (A fragment shared).
// Double-buffered: chunk k+1 staged via GLOBAL_LOAD_ASYNC_TO_LDS (b128 for A,
// b32 scatter for the transposed W tiles) while chunk k runs on the WMMAs.
// ---------------------------------------------------------------------------
#define APITCH 36                 // mult of 4 -> 16B-aligned b128 LDS rows
#define CHUNK_LDS (32*APITCH)     // floats per tile buffer
__global__ __launch_bounds__(128) void proj_qk_wmma(
    const float* __restrict__ query, const float* __restrict__ Wq,
    const float* __restrict__ bq,    const float* __restrict__ Wk,
    const float* __restrict__ bk,    float* __restrict__ Qws,
    float* __restrict__ Kout)
{
  __shared__ float sA [2*CHUNK_LDS];   // [buf][m][k]
  __shared__ float sBq[2*CHUNK_LDS];   // [buf][n][k]  (W transposed)
  __shared__ float sBk[2*CHUNK_LDS];
  const int t    = threadIdx.x;
  const int lane = t & 31;
  const int w    = t >> 5;
  const int wm   = w >> 1, wn = w & 1;
  const int m0   = blockIdx.y * 32;
  const int c0   = blockIdx.x * 32;
  const int mrow = lane & 15;
  const int ksel = lane >> 4;          // half-wave selector per ISA layout
  v8f accq = {}; v8f acck = {};

  auto issue_chunk = [&](int k0, int buf) {   // 18 async ops per thread
    #pragma unroll
    for (int s = 0; s < 2; ++s) {             // A tile: contiguous b128
      int idx4 = t + s*128;
      int rr = idx4 >> 3, cc4 = (idx4 & 7) * 4;
      async_ld_b128(lds_addr(&sA[buf*CHUNK_LDS + rr*APITCH + cc4]),
                    &query[(size_t)(m0+rr)*DMODEL + k0 + cc4]);
    }
    #pragma unroll
    for (int s = 0; s < 8; ++s) {             // W tiles: transpose scatter b32
      int idx = t + s*128;
      int rr = idx >> 5, cc = idx & 31;
      async_ld_b32(lds_addr(&sBq[buf*CHUNK_LDS + cc*APITCH + rr]),
                   &Wq[(size_t)(k0+rr)*NOUT + c0 + cc]);
      async_ld_b32(lds_addr(&sBk[buf*CHUNK_LDS + cc*APITCH + rr]),
                   &Wk[(size_t)(k0+rr)*NOUT + c0 + cc]);
    }
  };

  issue_chunk(0, 0);
  for (int kc = 0; kc < DMODEL/32; ++kc) {
    const int cur = kc & 1;
    if (kc < DMODEL/32 - 1) {
      issue_chunk((kc+1)*32, 1-cur);
      asm volatile("s_wait_asynccnt 0x12" ::: "memory");  // drain chunk kc (18)
    } else {
      asm volatile("s_wait_asynccnt 0x0" ::: "memory");
    }
    __syncthreads();
    const float* pa0  = &sA [cur*CHUNK_LDS + (wm*16 + mrow)*APITCH + ksel*2];
    const float* pbq0 = &sBq[cur*CHUNK_LDS + (wn*16 + mrow)*APITCH + ksel*2];
    const float* pbk0 = &sBk[cur*CHUNK_LDS + (wn*16 + mrow)*APITCH + ksel*2];
    #pragma unroll
    for (int k4 = 0; k4 < 8; ++k4) {          // eight K=4 WMMA steps
      v2f af  = *(const v2f*)(pa0  + k4*4);
      v2f bqf = *(const v2f*)(pbq0 + k4*4);
      v2f bkf = *(const v2f*)(pbk0 + k4*4);
      accq = __builtin_amdgcn_wmma_f32_16x16x4_f32(false, af, false, bqf,
                                                   (short)0, accq, false, false);
      acck = __builtin_amdgcn_wmma_f32_16x16x4_f32(false, af, false, bkf,
                                                   (short)0, acck, false, false);
    }
    __syncthreads();   // protect buffer 'cur' before chunk kc+2 overwrites it
  }
  // C/D layout: vgpr i -> (m = ksel*8+i, n = lane&15)
  const int nn = lane & 15;
  const int c  = c0 + wn*16 + nn;
  const int h  = c >> 6, e = c & 63;
  const float bqv = bq[c], bkv = bk[c];
  #pragma unroll
  for (int i = 0; i < 8; ++i) {
    int r = m0 + wm*16 + ksel*8 + i;    // fused row = s*B + b
    int s = r >> 1, b = r & 1;
    size_t off = ((size_t)((b*NHH + h)*NSQ + s))*DHH + e;
    Qws [off] = accq[i] + bqv;
    Kout[off] = acck[i] + bkv;          // k_out layout == [(b*8+h)][s][e]
  }
}

// ---------------------------------------------------------------------------
// Kernel B: Qp[bh,s] = <q[bh,s,:], p[h,:]>, Kp likewise (rank-1 leaky term).
// ---------------------------------------------------------------------------
__global__ __launch_bounds__(256) void dot_p(
    const float* __restrict__ Qws, const float* __restrict__ Kout,
    const float* __restrict__ proj, float* __restrict__ Qp,
    float* __restrict__ Kp)
{
  int idx = blockIdx.x*256 + threadIdx.x;   // 16384 threads
  int which = idx >> 13;
  int v  = idx & 8191;
  int bh = v >> 9;
  const float* src = (which ? Kout : Qws) + (size_t)v*DHH;
  const float* p   = proj + (bh & 7)*DHH;
  float sum = 0.f;
  #pragma unroll
  for (int e = 0; e < DHH; e += 4) {
    float4 a = *(const float4*)(src + e);
    float4 b = *(const float4*)(p + e);
    sum += a.x*b.x + a.y*b.y + a.z*b.z + a.w*b.w;
  }
  if (which) Kp[v] = sum; else Qp[v] = sum;
}

// ---------------------------------------------------------------------------
// Kernel C: attn[bh,i,j] = 0.505*(Qp[i]+Kp[j]) + 0.495*sum_e |q+k| * p[e]
// (leaky(x) = 0.505x + 0.495|x| exactly, slope 0.01; |x| folds into the
// VOP3 source-abs modifier of v_fma_f32 -> 2 VALU ops per inner element.)
// 64x64 tile staged with async b128 copies; thread -> 4x4 outputs with
// i = ty+16*ii, j = tx+16*jj (strided => conflict-free ds_load_b128).
// ---------------------------------------------------------------------------
#define CP 68
__global__ __launch_bounds__(256) void attn_kernel(
    const float* __restrict__ Qws, const float* __restrict__ Kout,
    const float* __restrict__ proj, const float* __restrict__ Qp,
    const float* __restrict__ Kp, float* __restrict__ attn)
{
  __shared__ float qs[64*CP];
  __shared__ float ks[64*CP];
  __shared__ float pl[DHH];
  __shared__ float qpl[64];
  __shared__ float kpl[64];
  const int t  = threadIdx.x;
  const int bh = blockIdx.z;
  const int i0 = blockIdx.y * 64;
  const int j0 = blockIdx.x * 64;
  const float* Qb = Qws  + ((size_t)bh*NSQ + i0)*DHH;   // contiguous 64x64 tile
  const float* Kb = Kout + ((size_t)bh*NSQ + j0)*DHH;
  #pragma unroll
  for (int s = 0; s < 4; ++s) {
    int idx4 = t + s*256;
    int row = idx4 >> 4;
    int col = (idx4 & 15) * 4;
    async_ld_b128(lds_addr(&qs[row*CP + col]), Qb + (size_t)idx4*4);
    async_ld_b128(lds_addr(&ks[row*CP + col]), Kb + (size_t)idx4*4);
  }
  if (t < 64) {
    pl[t]  = proj[(bh & 7)*DHH + t];
    qpl[t] = Qp[bh*NSQ + i0 + t];
    kpl[t] = Kp[bh*NSQ + j0 + t];
  }
  asm volatile("s_wait_asynccnt 0x0" ::: "memory");
  __syncthreads();
  const int tx = t & 15, ty = t >> 4;
  float acc[4][4] = {};
  #pragma unroll 4
  for (int e0 = 0; e0 < DHH; e0 += 4) {
    float4 p4 = *(const float4*)&pl[e0];
    float4 qv[4], kv[4];
    #pragma unroll
    for (int ii = 0; ii < 4; ++ii) qv[ii] = *(const float4*)&qs[(ty + ii*16)*CP + e0];
    #pragma unroll
    for (int jj = 0; jj < 4; ++jj) kv[jj] = *(const float4*)&ks[(tx + jj*16)*CP + e0];
    #pragma unroll
    for (int ii = 0; ii < 4; ++ii)
      #pragma unroll
      for (int jj = 0; jj < 4; ++jj) {
        acc[ii][jj] = __builtin_fmaf(__builtin_fabsf(qv[ii].x + kv[jj].x), p4.x, acc[ii][jj]);
        acc[ii][jj] = __builtin_fmaf(__builtin_fabsf(qv[ii].y + kv[jj].y), p4.y, acc[ii][jj]);
        acc[ii][jj] = __builtin_fmaf(__builtin_fabsf(qv[ii].z + kv[jj].z), p4.z, acc[ii][jj]);
        acc[ii][jj] = __builtin_fmaf(__builtin_fabsf(qv[ii].w + kv[jj].w), p4.w, acc[ii][jj]);
      }
  }
  float* ob = attn + ((size_t)bh*NSQ + i0)*NSQ + j0;
  #pragma unroll
  for (int ii = 0; ii < 4; ++ii) {
    int i = ty + ii*16;
    float qc = 0.505f * qpl[i];
    #pragma unroll
    for (int jj = 0; jj < 4; ++jj) {
      int j = tx + jj*16;
      ob[(size_t)i*NSQ + j] = 0.495f*acc[ii][jj] + qc + 0.505f*kpl[j];
    }
  }
}

extern "C" void kernel_launch(void* const* d_in, const int* in_sizes, int n_in,
                              void* d_out, int out_size, void* d_ws, size_t ws_size,
                              hipStream_t stream) {
  const float* query = (const float*)d_in[0];
  const float* Wq    = (const float*)d_in[1];
  const float* bq    = (const float*)d_in[2];
  const float* Wk    = (const float*)d_in[3];
  const float* bk    = (const float*)d_in[4];
  const float* proj  = (const float*)d_in[5];

  float* attn = (float*)d_out;
  float* kout = (float*)d_out + ATTN_ELEMS;     // k_out written in final layout

  float* Qws = (float*)d_ws;                    // 524288 floats
  float* Qp  = Qws + KOUT_ELEMS;                // 8192 floats
  float* Kp  = Qp  + BB*NHH*NSQ;                // 8192 floats  (~2.2 MB total)

  proj_qk_wmma<<<dim3(NOUT/32, NSB/32), 128, 0, stream>>>(query, Wq, bq, Wk, bk,
                                                          Qws, kout);
  dot_p<<<64, 256, 0, stream>>>(Qws, kout, proj, Qp, Kp);
  attn_kernel<<<dim3(NSQ/64, NSQ/64, BB*NHH), 256, 0, stream>>>(Qws, kout, proj,
                                                                Qp, Kp, attn);
}